// GCN1_63024350101691
// MI455X (gfx1250) — compile-verified
//
#include <hip/hip_runtime.h>
#include <hip/hip_bf16.h>

#define N_NODES 50000
#define N_EDGES 800000
#define D 128
#define OUT_DIM 10
#define EPS 1e-5f
#define ROWS_PER_BLOCK 128

typedef float v2f __attribute__((ext_vector_type(2)));
typedef float v8f __attribute__((ext_vector_type(8)));

// ---------------- edge scatter: agg[dst] += x[src] -------------------------
// one wave (32 lanes) per edge; each lane moves 4 consecutive floats (coalesced
// 512B gather per edge) and lands them with hardware f32 atomics (L2-resident).
__global__ void scatter_kernel(const long long* __restrict__ srcIdx,
                               const long long* __restrict__ dstIdx,
                               const float* __restrict__ x,
                               float* __restrict__ agg) {
  unsigned long long t = (unsigned long long)blockIdx.x * blockDim.x + threadIdx.x;
  unsigned e = (unsigned)(t >> 5);
  unsigned c = (unsigned)(t & 31u) * 4u;
  if (e >= N_EDGES) return;
  long long s = srcIdx[e];
  long long d = dstIdx[e];
  const float4 v = *(const float4*)(x + s * D + c);
  float* o = agg + d * D + c;
  unsafeAtomicAdd(o + 0, v.x);
  unsafeAtomicAdd(o + 1, v.y);
  unsafeAtomicAdd(o + 2, v.z);
  unsafeAtomicAdd(o + 3, v.w);
}

// ---------------- fused GraphConv GEMM -------------------------------------
// out = agg @ Wrel^T + x @ Wroot^T + b     (all [N,128] x [128,128])
// block = 256 threads = 8 waves; block owns 16 rows, wave w owns cols [16w,16w+16)
// V_WMMA_F32_16X16X4_F32, K stepped by 4, full f32 precision.
__global__ void __launch_bounds__(256)
gemm_conv_kernel(const float* __restrict__ agg, const float* __restrict__ x,
                 const float* __restrict__ Wrel, const float* __restrict__ Wroot,
                 const float* __restrict__ bias, float* __restrict__ out) {
  const int lane = threadIdx.x & 31;
  const int wave = threadIdx.x >> 5;        // column tile 0..7
  const int m    = lane & 15;
  const int hi   = lane >> 4;               // lane-half selects K pair / M half
  const int kh   = hi * 2;
  const int rowTile = blockIdx.x;           // 0..3124  (50000 = 3125*16)
  const int row  = rowTile * 16 + m;
  const int n    = wave * 16 + m;           // B column index for this lane
  const float* aR = agg  + (size_t)row * D;
  const float* xR = x    + (size_t)row * D;
  const float* bR = Wrel  + (size_t)n * D;  // B[k][n] = Wrel[n*D + k]
  const float* bT = Wroot + (size_t)n * D;

  v8f c = {0.f, 0.f, 0.f, 0.f, 0.f, 0.f, 0.f, 0.f};
#pragma unroll 4
  for (int k0 = 0; k0 < D; k0 += 4) {
    v2f a = *(const v2f*)(aR + k0 + kh);
    v2f b = *(const v2f*)(bR + k0 + kh);
    c = __builtin_amdgcn_wmma_f32_16x16x4_f32(false, a, false, b, (short)0, c,
                                              false, false);
  }
#pragma unroll 4
  for (int k0 = 0; k0 < D; k0 += 4) {
    v2f a = *(const v2f*)(xR + k0 + kh);
    v2f b = *(const v2f*)(bT + k0 + kh);
    c = __builtin_amdgcn_wmma_f32_16x16x4_f32(false, a, false, b, (short)0, c,
                                              false, false);
  }
  const float bn = bias[n];
  const int mbase = rowTile * 16 + hi * 8;  // C/D layout: VGPR r -> row r + 8*hi
#pragma unroll
  for (int r = 0; r < 8; ++r)
    out[(size_t)(mbase + r) * D + n] = c[r] + bn;
}

// ---------------- GraphNorm stats + apply ----------------------------------
__global__ void colsum_kernel(const float* __restrict__ x, float* __restrict__ sums) {
  int t = threadIdx.x;                       // feature 0..127
  int r0 = blockIdx.x * ROWS_PER_BLOCK;
  int r1 = r0 + ROWS_PER_BLOCK; if (r1 > N_NODES) r1 = N_NODES;
  float acc = 0.f;
  for (int r = r0; r < r1; ++r) acc += x[(size_t)r * D + t];
  unsafeAtomicAdd(&sums[t], acc);
}

__global__ void colvar_kernel(const float* __restrict__ x, const float* __restrict__ sums,
                              const float* __restrict__ alpha, float* __restrict__ vars) {
  int t = threadIdx.x;
  float am = alpha[t] * (sums[t] * (1.f / N_NODES));
  int r0 = blockIdx.x * ROWS_PER_BLOCK;
  int r1 = r0 + ROWS_PER_BLOCK; if (r1 > N_NODES) r1 = N_NODES;
  float acc = 0.f;
  for (int r = r0; r < r1; ++r) { float v = x[(size_t)r * D + t] - am; acc += v * v; }
  unsafeAtomicAdd(&vars[t], acc);
}

__global__ void norm_leaky_kernel(const float* __restrict__ x, float* __restrict__ y,
                                  const float* __restrict__ sums, const float* __restrict__ vars,
                                  const float* __restrict__ gamma, const float* __restrict__ beta,
                                  const float* __restrict__ alpha) {
  int t = threadIdx.x;
  float am = alpha[t] * (sums[t] * (1.f / N_NODES));
  float sc = gamma[t] * rsqrtf(vars[t] * (1.f / N_NODES) + EPS);
  float bt = beta[t];
  int r0 = blockIdx.x * ROWS_PER_BLOCK;
  int r1 = r0 + ROWS_PER_BLOCK; if (r1 > N_NODES) r1 = N_NODES;
  for (int r = r0; r < r1; ++r) {
    float v = (x[(size_t)r * D + t] - am) * sc + bt;
    y[(size_t)r * D + t] = v >= 0.f ? v : 0.1f * v;
  }
}

// final layer: h = relu(feat + norm(x)); only the column sum of h is needed.
__global__ void norm_res_pool_kernel(const float* __restrict__ x, const float* __restrict__ feat,
                                     const float* __restrict__ sums, const float* __restrict__ vars,
                                     const float* __restrict__ gamma, const float* __restrict__ beta,
                                     const float* __restrict__ alpha, float* __restrict__ pooled) {
  int t = threadIdx.x;
  float am = alpha[t] * (sums[t] * (1.f / N_NODES));
  float sc = gamma[t] * rsqrtf(vars[t] * (1.f / N_NODES) + EPS);
  float bt = beta[t];
  int r0 = blockIdx.x * ROWS_PER_BLOCK;
  int r1 = r0 + ROWS_PER_BLOCK; if (r1 > N_NODES) r1 = N_NODES;
  float acc = 0.f;
  for (int r = r0; r < r1; ++r) {
    float h = feat[(size_t)r * D + t] + (x[(size_t)r * D + t] - am) * sc + bt;
    acc += fmaxf(h, 0.f);
  }
  unsafeAtomicAdd(&pooled[t], acc);
}

// ---------------- output head ----------------------------------------------
__global__ void head_kernel(const float* __restrict__ pooled, const float* __restrict__ Wlin,
                            const float* __restrict__ blin, float* __restrict__ out) {
  int j = threadIdx.x;
  if (j < OUT_DIM) {
    float s = blin[j];
    for (int k = 0; k < D; ++k) s += (pooled[k] * (1.f / N_NODES)) * Wlin[j * D + k];
    out[j] = fmaxf(s, 0.f);
  }
}

extern "C" void kernel_launch(void* const* d_in, const int* in_sizes, int n_in,
                              void* d_out, int out_size, void* d_ws, size_t ws_size,
                              hipStream_t stream) {
  const long long* edge   = (const long long*)d_in[0];  // int64 [2, E]
  const long long* srcIdx = edge;
  const long long* dstIdx = edge + N_EDGES;
  const float* feat  = (const float*)d_in[1];
  const float* Wrel  = (const float*)d_in[2];
  const float* Wroot = (const float*)d_in[3];
  const float* bconv = (const float*)d_in[4];
  const float* gamma = (const float*)d_in[5];
  const float* beta  = (const float*)d_in[6];
  const float* alpha = (const float*)d_in[7];
  const float* Wlin  = (const float*)d_in[8];
  const float* blin  = (const float*)d_in[9];
  float* outp = (float*)d_out;

  float* bufA  = (float*)d_ws;
  float* bufB  = bufA + (size_t)N_NODES * D;
  float* agg   = bufB + (size_t)N_NODES * D;
  float* stats = agg  + (size_t)N_NODES * D;
  float* sums = stats; float* vars = stats + D; float* pooled = stats + 2 * D;

  const int scatterBlocks = (N_EDGES * 32 + 255) / 256;                 // 100000
  const int gemmBlocks    = N_NODES / 16;                               // 3125
  const int colBlocks     = (N_NODES + ROWS_PER_BLOCK - 1) / ROWS_PER_BLOCK;

  const float* hin = feat;
  float* gemmOut[4] = {bufA, bufB, bufA, bufB};

  for (int l = 0; l < 4; ++l) {
    hipMemsetAsync(agg, 0, (size_t)N_NODES * D * sizeof(float), stream);
    hipMemsetAsync(stats, 0, (size_t)3 * D * sizeof(float), stream);
    scatter_kernel<<<scatterBlocks, 256, 0, stream>>>(srcIdx, dstIdx, hin, agg);
    float* ho = gemmOut[l];
    gemm_conv_kernel<<<gemmBlocks, 256, 0, stream>>>(
        agg, hin, Wrel + (size_t)l * D * D, Wroot + (size_t)l * D * D,
        bconv + (size_t)l * D, ho);
    colsum_kernel<<<colBlocks, D, 0, stream>>>(ho, sums);
    colvar_kernel<<<colBlocks, D, 0, stream>>>(ho, sums, alpha + (size_t)l * D, vars);
    if (l < 3) {
      norm_leaky_kernel<<<colBlocks, D, 0, stream>>>(ho, ho, sums, vars,
                                                     gamma + (size_t)l * D,
                                                     beta + (size_t)l * D,
                                                     alpha + (size_t)l * D);
      hin = ho;
    } else {
      norm_res_pool_kernel<<<colBlocks, D, 0, stream>>>(ho, feat, sums, vars,
                                                        gamma + 3 * D, beta + 3 * D,
                                                        alpha + 3 * D, pooled);
    }
  }
  head_kernel<<<1, 64, 0, stream>>>(pooled, Wlin, blin, outp);
}